// SCoNeLayer_20151986553302
// MI455X (gfx1250) — compile-verified
//
#include <hip/hip_runtime.h>
#include <hip/hip_bf16.h>

typedef __attribute__((ext_vector_type(2))) float v2f;
typedef __attribute__((ext_vector_type(8))) float v8f;

#define NE 1000000
#define NN 500000
#define NTRI 600000
#define CH 64
#define WPB 8            // waves per block in fused kernel
#define FUSED_BLOCKS 1536

// ---------------- zero workspace ----------------
__global__ void zero_ws_kernel(float4* __restrict__ p, int n4) {
    int i = blockIdx.x * blockDim.x + threadIdx.x;
    if (i < n4) p[i] = make_float4(0.f, 0.f, 0.f, 0.f);
}

// ---------------- y = scatter(+/- x) over edge endpoints ----------------
// one thread per (edge, channel-pair): float2 load, 4 f32 atomics
__global__ void scatter_y_kernel(const float* __restrict__ x,
                                 const long long* __restrict__ en,
                                 float* __restrict__ y) {
    int gid = blockIdx.x * blockDim.x + threadIdx.x;
    int e  = gid >> 5;
    int c  = (gid & 31) << 1;
    if (e >= NE) return;
    float2 v = *(const float2*)(x + (size_t)e * CH + c);
    long long tl = en[2 * (size_t)e + 0];
    long long hd = en[2 * (size_t)e + 1];
    float* yt = y + (size_t)tl * CH + c;
    float* yh = y + (size_t)hd * CH + c;
    atomicAdd(yt + 0, -v.x);
    atomicAdd(yt + 1, -v.y);
    atomicAdd(yh + 0,  v.x);
    atomicAdd(yh + 1,  v.y);
}

// ---------------- z1e = scatter of w = x[e0]-x[e1]+x[e2] ----------------
__global__ void scatter_tri_kernel(const float* __restrict__ x,
                                   const long long* __restrict__ te,
                                   float* __restrict__ z1e) {
    int gid = blockIdx.x * blockDim.x + threadIdx.x;
    int t = gid >> 5;
    int c = (gid & 31) << 1;
    if (t >= NTRI) return;
    long long e0 = te[3 * (size_t)t + 0];
    long long e1 = te[3 * (size_t)t + 1];
    long long e2 = te[3 * (size_t)t + 2];
    float2 a = *(const float2*)(x + (size_t)e0 * CH + c);
    float2 b = *(const float2*)(x + (size_t)e1 * CH + c);
    float2 d = *(const float2*)(x + (size_t)e2 * CH + c);
    float2 w = make_float2(a.x - b.x + d.x, a.y - b.y + d.y);
    float* p0 = z1e + (size_t)e0 * CH + c;
    float* p1 = z1e + (size_t)e1 * CH + c;
    float* p2 = z1e + (size_t)e2 * CH + c;
    atomicAdd(p0 + 0,  w.x);
    atomicAdd(p0 + 1,  w.y);
    atomicAdd(p1 + 0, -w.x);
    atomicAdd(p1 + 1, -w.y);
    atomicAdd(p2 + 0,  w.x);
    atomicAdd(p2 + 1,  w.y);
}

// ---------------- fused: out = tanh(z1e@w2 + x@w1 + (y[h]-y[t])@w0) ----------------
// Persistent waves: grid-stride over 16-edge tiles. f32 WMMA 16x16x4, K swept in
// 16 steps over C=64, 4 n-tiles of 16 columns. Weights transposed in LDS once per
// block so B-fragments are contiguous b64 LDS loads. Next tile's rows prefetched.
__global__ void __launch_bounds__(WPB * 32)
fused_wmma_kernel(const float* __restrict__ x,
                  const long long* __restrict__ en,
                  const float* __restrict__ w0,
                  const float* __restrict__ w1,
                  const float* __restrict__ w2,
                  const float* __restrict__ y,
                  const float* __restrict__ z1e,
                  float* __restrict__ out,
                  int ntiles) {
    __shared__ float wt[3][CH * CH];   // transposed: wt[s][n*64 + k] = ws[k*64 + n]

    for (int i = threadIdx.x; i < CH * CH; i += blockDim.x) {
        int k = i >> 6;
        int n = i & 63;
        wt[0][n * CH + k] = w0[i];
        wt[1][n * CH + k] = w1[i];
        wt[2][n * CH + k] = w2[i];
    }
    __syncthreads();

    const int wave = threadIdx.x >> 5;
    const int lane = threadIdx.x & 31;
    const int m    = lane & 15;            // A: row M ; B: col N (within n-tile)
    const int kb   = (lane >> 4) << 1;     // K offset 0 (lanes 0-15) or 2 (lanes 16-31)
    const int nwaves = gridDim.x * WPB;

    for (int tile = blockIdx.x * WPB + wave; tile < ntiles; tile += nwaves) {
        int e = tile * 16 + m;
        long long tl = en[2 * (size_t)e + 0];
        long long hd = en[2 * (size_t)e + 1];
        const float* xr = x   + (size_t)e  * CH;
        const float* zr = z1e + (size_t)e  * CH;
        const float* yh = y   + (size_t)hd * CH;
        const float* yt = y   + (size_t)tl * CH;

        // Prefetch next tile's rows (global_prefetch_b8); hides gather latency.
        int tn = tile + nwaves;
        if (tn < ntiles) {
            size_t e2 = (size_t)(tn * 16 + m);
            __builtin_prefetch(x   + e2 * CH,      0, 3);
            __builtin_prefetch(x   + e2 * CH + 32, 0, 3);
            __builtin_prefetch(z1e + e2 * CH,      0, 3);
            __builtin_prefetch(z1e + e2 * CH + 32, 0, 3);
            long long tl2 = en[2 * e2 + 0];
            long long hd2 = en[2 * e2 + 1];
            __builtin_prefetch(y + (size_t)tl2 * CH,      0, 3);
            __builtin_prefetch(y + (size_t)tl2 * CH + 32, 0, 3);
            __builtin_prefetch(y + (size_t)hd2 * CH,      0, 3);
            __builtin_prefetch(y + (size_t)hd2 * CH + 32, 0, 3);
        }

        v8f acc[4] = {};

        for (int kk = 0; kk < 16; ++kk) {
            int col = (kk << 2) + kb;      // K pair start within C=64
            v2f ax  = *(const v2f*)(xr + col);
            v2f az  = *(const v2f*)(zr + col);
            v2f ayh = *(const v2f*)(yh + col);
            v2f ayt = *(const v2f*)(yt + col);
            v2f ay  = ayh - ayt;
#pragma unroll
            for (int nt = 0; nt < 4; ++nt) {
                int nb = (nt * 16 + m) * CH + col;
                v2f b0 = *(const v2f*)(&wt[0][nb]);
                v2f b1 = *(const v2f*)(&wt[1][nb]);
                v2f b2 = *(const v2f*)(&wt[2][nb]);
                acc[nt] = __builtin_amdgcn_wmma_f32_16x16x4_f32(
                    false, ay, false, b0, (short)0, acc[nt], false, false);
                acc[nt] = __builtin_amdgcn_wmma_f32_16x16x4_f32(
                    false, ax, false, b1, (short)0, acc[nt], false, false);
                acc[nt] = __builtin_amdgcn_wmma_f32_16x16x4_f32(
                    false, az, false, b2, (short)0, acc[nt], false, false);
            }
        }

        // C/D layout: VGPR i -> row i (lanes 0-15) / row i+8 (lanes 16-31); col = lane&15
        int rbase = tile * 16 + ((lane >> 4) << 3);
#pragma unroll
        for (int i = 0; i < 8; ++i) {
            float* orow = out + (size_t)(rbase + i) * CH + m;
#pragma unroll
            for (int nt = 0; nt < 4; ++nt)
                orow[nt * 16] = tanhf(acc[nt][i]);
        }
    }
}

extern "C" void kernel_launch(void* const* d_in, const int* in_sizes, int n_in,
                              void* d_out, int out_size, void* d_ws, size_t ws_size,
                              hipStream_t stream) {
    const float*     x  = (const float*)d_in[0];
    const long long* en = (const long long*)d_in[1];   // edge_nodes int64 (N_EDGES,2)
    const long long* te = (const long long*)d_in[2];   // tri_edges  int64 (N_TRIS,3)
    const float*     w0 = (const float*)d_in[3];
    const float*     w1 = (const float*)d_in[4];
    const float*     w2 = (const float*)d_in[5];
    float* out = (float*)d_out;

    float* y   = (float*)d_ws;                  // N_NODES*64 f32 = 128 MB
    float* z1e = y + (size_t)NN * CH;           // N_EDGES*64 f32 = 256 MB

    int n4 = (NN + NE) * CH / 4;                // 24M float4
    zero_ws_kernel<<<(n4 + 255) / 256, 256, 0, stream>>>((float4*)d_ws, n4);

    scatter_y_kernel<<<(NE * 32 + 255) / 256, 256, 0, stream>>>(x, en, y);
    scatter_tri_kernel<<<(NTRI * 32 + 255) / 256, 256, 0, stream>>>(x, te, z1e);

    int ntiles = NE / 16;                       // 62500
    int blocks = (ntiles + WPB - 1) / WPB;
    if (blocks > FUSED_BLOCKS) blocks = FUSED_BLOCKS;
    fused_wmma_kernel<<<blocks, WPB * 32, 0, stream>>>(
        x, en, w0, w1, w2, y, z1e, out, ntiles);
}